// GSOrthogonal_36180804501946
// MI455X (gfx1250) — compile-verified
//
#include <hip/hip_runtime.h>

typedef __attribute__((ext_vector_type(2))) float v2f;
typedef __attribute__((ext_vector_type(8))) float v8f;

#define N_FULL   4096
#define BS       64
#define NBLK     64
#define BROWS    16
#define LSTRIDE  65                      // padded per-l sub-block stride (floats)
#define RS       (NBLK*LSTRIDE + 4)     // 4164 floats per mid row (4164 % 64 == 4)

static __device__ inline v8f wmma4(v2f a, v2f b, v8f c) {
  // V_WMMA_F32_16X16X4_F32: D = A(16x4) * B(4x16) + C
  return __builtin_amdgcn_wmma_f32_16x16x4_f32(false, a, false, b, (short)0, c,
                                               false, false);
}

// ---------------------------------------------------------------------------
// Kernel 1: Cayley orthogonalization. One 64x64 matrix per workgroup.
// Q = (I + S) * inv(I - S), S = 0.5*(M - M^T). (I-S) is well-conditioned
// (eigenvalues 1 - i*lambda), so Gauss-Jordan without pivoting is safe.
// ---------------------------------------------------------------------------
__global__ __launch_bounds__(64)
void cayley_kernel(const float* __restrict__ L, const float* __restrict__ R,
                   float* __restrict__ Lq, float* __restrict__ Rq) {
  __shared__ float S[64][66];
  __shared__ float W[64][66];   // holds A = I - S; Gauss-Jordan reduces it to I
  __shared__ float E[64][66];   // identity -> inv(A)
  const int mat = blockIdx.x;                       // 0..127
  const float* M = (mat < 64) ? (L + (size_t)mat * 4096)
                              : (R + (size_t)(mat - 64) * 4096);
  float* Q = (mat < 64) ? (Lq + (size_t)mat * 4096)
                        : (Rq + (size_t)(mat - 64) * 4096);
  const int t = threadIdx.x;                        // 0..63

  for (int idx = t; idx < 4096; idx += 64) {
    const int i = idx >> 6, j = idx & 63;
    const float s = 0.5f * (M[i * 64 + j] - M[j * 64 + i]);
    const float d = (i == j) ? 1.0f : 0.0f;
    S[i][j] = s;
    W[i][j] = d - s;            // A = I - S
    E[i][j] = d;
  }
  __syncthreads();

  // Gauss-Jordan: W -> I, E -> inv(A)
  for (int p = 0; p < 64; ++p) {
    const float invp = 1.0f / W[p][p];
    __syncthreads();
    // thread t scales column t of the pivot row
    W[p][t] *= invp;
    E[p][t] *= invp;
    __syncthreads();
    if (t != p) {
      const float f = W[t][p];
      #pragma unroll 8
      for (int j = 0; j < 64; ++j) {
        W[t][j] = fmaf(-f, W[p][j], W[t][j]);
        E[t][j] = fmaf(-f, E[p][j], E[t][j]);
      }
    }
    __syncthreads();
  }

  // Q = (I + S) * inv(A); thread t computes row t
  for (int j = 0; j < 64; ++j) {
    float acc = E[t][j];
    #pragma unroll 8
    for (int k = 0; k < 64; ++k) acc = fmaf(S[t][k], E[k][j], acc);
    Q[t * 64 + j] = acc;
  }
}

// ---------------------------------------------------------------------------
// Kernel 2: fused block-diag butterfly with fp32 WMMA.
// out[b, s*64+l] = sum_r Lq[l,s,r] * ( sum_p x[b, r*64+p] * Rq[r,l,p] )
// Intermediate (16 x 4096 per WG) lives entirely in LDS.
// ---------------------------------------------------------------------------
__global__ __launch_bounds__(256, 1)
void butterfly_wmma_kernel(const float* __restrict__ x,
                           const float* __restrict__ Rq,
                           const float* __restrict__ Lq,
                           float* __restrict__ out) {
  __shared__ float mid[BROWS * RS];     // ~260 KB (<= 320 KB/WGP)

  const int tid  = threadIdx.x;
  const int wave = tid >> 5;            // 0..7
  const int lane = tid & 31;
  const int hi   = lane >> 4;           // 0: K pair k0,k0+1 ; 1: K pair k0+2,k0+3
  const int lq   = lane & 15;           // A: M row / B,D: N column
  const long b0  = (long)blockIdx.x * BROWS;

  // ---- Stage 1: for each r-block: mid(.,l-slice r) = X_r (16x64) @ Rq[r]^T
  const float* xr_base = x + (b0 + lq) * (long)N_FULL + 2 * hi;
  for (int r = wave; r < NBLK; r += 8) {
    const float* xr = xr_base + r * BS;
    const float* w  = Rq + (size_t)r * (BS * BS) + lq * BS + 2 * hi;
    v8f c0 = {}, c1 = {}, c2 = {}, c3 = {};
    #pragma unroll
    for (int k0 = 0; k0 < BS; k0 += 4) {
      const v2f av = *(const v2f*)(xr + k0);          // A[m=lq][k0+2*hi .. +1]
      const v2f b0v = *(const v2f*)(w + k0);          // B[k][n], n = lq
      const v2f b1v = *(const v2f*)(w + 1024 + k0);   // n = 16 + lq
      const v2f b2v = *(const v2f*)(w + 2048 + k0);   // n = 32 + lq
      const v2f b3v = *(const v2f*)(w + 3072 + k0);   // n = 48 + lq
      c0 = wmma4(av, b0v, c0);
      c1 = wmma4(av, b1v, c1);
      c2 = wmma4(av, b2v, c2);
      c3 = wmma4(av, b3v, c3);
    }
    // D layout: lane<16 -> D[m=v][n=lq], lane>=16 -> D[m=v+8][n=lq]
    #pragma unroll
    for (int v = 0; v < 8; ++v) {
      const int m = v + 8 * hi;
      float* mrow = &mid[m * RS + r];
      mrow[(lq)      * LSTRIDE] = c0[v];
      mrow[(16 + lq) * LSTRIDE] = c1[v];
      mrow[(32 + lq) * LSTRIDE] = c2[v];
      mrow[(48 + lq) * LSTRIDE] = c3[v];
    }
  }
  __syncthreads();

  // ---- Stage 2: for each l: Y_l = T_l (16x64) @ Lq[l]^T, T_l[m,r] in LDS.
  // Result overwrites the (wave-private, already consumed) l-slice as
  // mid[m][l*65 + s].
  for (int l = wave; l < NBLK; l += 8) {
    const float* w  = Lq + (size_t)l * (BS * BS) + lq * BS + 2 * hi;
    const float* am = &mid[lq * RS + l * LSTRIDE + 2 * hi];
    v8f c0 = {}, c1 = {}, c2 = {}, c3 = {};
    #pragma unroll
    for (int k0 = 0; k0 < BS; k0 += 4) {
      v2f av; av.x = am[k0]; av.y = am[k0 + 1];       // may be 4B-aligned only
      const v2f b0v = *(const v2f*)(w + k0);
      const v2f b1v = *(const v2f*)(w + 1024 + k0);
      const v2f b2v = *(const v2f*)(w + 2048 + k0);
      const v2f b3v = *(const v2f*)(w + 3072 + k0);
      c0 = wmma4(av, b0v, c0);
      c1 = wmma4(av, b1v, c1);
      c2 = wmma4(av, b2v, c2);
      c3 = wmma4(av, b3v, c3);
    }
    #pragma unroll
    for (int v = 0; v < 8; ++v) {
      const int m = v + 8 * hi;
      float* mrow = &mid[m * RS + l * LSTRIDE];
      mrow[lq]      = c0[v];   // s = lq
      mrow[16 + lq] = c1[v];
      mrow[32 + lq] = c2[v];
      mrow[48 + lq] = c3[v];
    }
  }
  __syncthreads();

  // ---- Final sweep: coalesced stores. out col c = s*64 + l, value at
  // mid[m][(c&63)*65 + (c>>6)]. Lane-consecutive c -> LDS bank stride 65
  // (conflict-free) and contiguous 128B global stores.
  for (int m = 0; m < BROWS; ++m) {
    float* orow = out + (b0 + m) * (long)N_FULL;
    const float* mrow = &mid[m * RS];
    for (int c = tid; c < N_FULL; c += 256) {
      orow[c] = mrow[(c & 63) * LSTRIDE + (c >> 6)];
    }
  }
}

extern "C" void kernel_launch(void* const* d_in, const int* in_sizes, int n_in,
                              void* d_out, int out_size, void* d_ws, size_t ws_size,
                              hipStream_t stream) {
  (void)n_in; (void)out_size; (void)ws_size;
  const float* x = (const float*)d_in[0];
  const float* L = (const float*)d_in[1];
  const float* R = (const float*)d_in[2];
  float* out = (float*)d_out;

  float* Lq = (float*)d_ws;                 // 64*64*64 floats = 1 MB
  float* Rq = Lq + 64 * 64 * 64;            // next 1 MB

  const int batch = in_sizes[0] / N_FULL;   // 16384

  cayley_kernel<<<128, 64, 0, stream>>>(L, R, Lq, Rq);
  butterfly_wmma_kernel<<<batch / BROWS, 256, 0, stream>>>(x, Rq, Lq, out);
}